// OrthogonalButterfly_41592463294681
// MI455X (gfx1250) — compile-verified
//
#include <hip/hip_runtime.h>

// CDNA5 / gfx1250 wave32 WMMA implementation of the 24-step orthogonal
// butterfly network (see round-1 derivation):
//   4 butterfly steps == one dense 16x16 orthogonal matrix per column group
//   {r + 256*m, m=0..15} -> {16r + t, t=0..15};  24 steps = 6 macro passes of
//   256 independent 16x16 matmuls per 16-row tile -> V_WMMA_F32_16X16X4_F32.
// Rows stay LDS-resident for all 6 passes: HBM sees exactly one 128 MB read
// and one 128 MB write (the roofline floor at 23.3 TB/s).
//
// This revision restructures all addressing so the compiler emits
// base-VGPR + immediate-offset loads (no per-group 64-bit address pools,
// which caused scratch spills in the previous revision):
//   * B (16x16 fused matrices) re-laid out as W[K][r][kk][lane][2] so each
//     lane fetches its two K-slice values with ONE global_load_b64 off a
//     single per-thread VGPR + immediate.
//   * A reads / D writes use one precomputed LDS base VGPR + ds immediates.

typedef __attribute__((ext_vector_type(2))) float v2f;
typedef __attribute__((ext_vector_type(8))) float v8f;

#define DEPTH        24
#define NCOL         4096
#define BATCH        8192
#define NGROUP       256       // 16-column groups per macro step
#define NMACRO       6         // 24 steps / 4 fused steps
#define ROWS_PER_WG  16
#define WAVES        16
#define THREADS      (WAVES * 32)
#define LDS_STRIDE   4100      // padded row stride (floats): bank spread, 16B aligned
#define WK_FLOATS    (NGROUP * 256)   // 65536 floats per macro step in W

// ---------------------------------------------------------------------------
// Prep: fused 4-step 16x16 matrices.  One thread per (K, r, basis m).
// Sub-step w pairs local slots (m, m+8); theta row index:
//   phi_w(m) = (r<<w) + ((m>>w)<<(8+w)) + (m & ((1<<w)-1)).
// Output layout (floats):  W[K*65536 + r*256 + kk*64 + (hi*16 + t)*2 + j]
// holding coefficient of input m = kk*4 + hi*2 + j for output t.
// ---------------------------------------------------------------------------
__global__ void butterfly_prep(const float* __restrict__ params,
                               float* __restrict__ W) {
  const int t  = blockIdx.x * blockDim.x + threadIdx.x;  // 6*256*16 threads
  const int b  = t & 15;          // basis column m
  const int gr = (t >> 4) & 255;  // group r
  const int K  = t >> 12;         // macro step
  if (K >= NMACRO) return;

  float v[16];
#pragma unroll
  for (int i = 0; i < 16; ++i) v[i] = (i == b) ? 1.0f : 0.0f;

#pragma unroll
  for (int w = 0; w < 4; ++w) {
    float nv[16];
#pragma unroll
    for (int m = 0; m < 8; ++m) {
      const int j = (gr << w) + ((m >> w) << (8 + w)) + (m & ((1 << w) - 1));
      const float th = params[j * DEPTH + (4 * K + w)];  // params: (2048, 24) row-major
      const float c = cosf(th);
      const float s = sinf(th);
      nv[2 * m]     =  c * v[m] + s * v[m + 8];
      nv[2 * m + 1] = -s * v[m] + c * v[m + 8];
    }
#pragma unroll
    for (int i = 0; i < 16; ++i) v[i] = nv[i];
  }

  // Element (m=b, t) -> kk = b>>2, hi = (b>>1)&1, j = b&1.
  const int kk = b >> 2;
  const int hi = (b >> 1) & 1;
  const int jj = b & 1;
  float* dst = W + (size_t)K * WK_FLOATS + gr * 256 + kk * 64 + hi * 32 + jj;
#pragma unroll
  for (int i = 0; i < 16; ++i) dst[i * 2] = v[i];
}

// ---------------------------------------------------------------------------
// Main kernel.
// ---------------------------------------------------------------------------
__global__ void __launch_bounds__(THREADS, 1)
butterfly_main(const float* __restrict__ X,
               const float* __restrict__ W,
               float* __restrict__ out) {
  extern __shared__ float lds[];  // ROWS_PER_WG * LDS_STRIDE floats (256.25 KB)

  const int tid  = threadIdx.x;
  const int lane = tid & 31;
  const int wave = tid >> 5;
  const size_t rowBase = (size_t)blockIdx.x * ROWS_PER_WG;

  // ---- load 16 rows (b128, fully coalesced) --------------------------------
  const float4* src = (const float4*)(X + rowBase * NCOL);
  for (int i = tid; i < ROWS_PER_WG * NCOL / 4; i += THREADS) {
    const int r  = i >> 10;          // NCOL/4 == 1024
    const int c4 = i & 1023;
    const float4 d = src[r * 1024 + c4];
    *(float4*)&lds[r * LDS_STRIDE + c4 * 4] = d;
  }
  __syncthreads();

  const int hi  = lane >> 4;   // 0: lanes 0-15, 1: lanes 16-31
  const int l16 = lane & 15;

  // Per-thread invariant bases (computed once; all inner accesses use
  // immediate offsets off these).
  const int abase = l16 * LDS_STRIDE + wave * 16 + hi * 512;   // A reads (floats)
  const int bvoff = wave * 4096 + lane * 2;                    // B reads  (floats)
  int dbase[8];
#pragma unroll
  for (int v = 0; v < 8; ++v)                                  // D writes (floats)
    dbase[v] = (v + hi * 8) * LDS_STRIDE + wave * 256 + l16;

  v8f acc[16];

  for (int K = 0; K < NMACRO; ++K) {
    const float* __restrict__ Wk = W + (size_t)K * WK_FLOATS;

    // ---- compute phase: all LDS reads of this pass happen here -------------
#pragma unroll
    for (int g = 0; g < 16; ++g) {
      v8f c = {0.f, 0.f, 0.f, 0.f, 0.f, 0.f, 0.f, 0.f};
#pragma unroll
      for (int kk = 0; kk < 4; ++kk) {
        // A (16x4 f32): M = l16; lanes<16 -> K = kk*4+{0,1}, lanes>=16 -> +{2,3}.
        v2f a;
        a.x = lds[abase + g + kk * 1024];        // col r + 256*m0
        a.y = lds[abase + g + kk * 1024 + 256];  // col r + 256*(m0+1)
        // B (4x16 f32): one b64 per lane, layout arranged by butterfly_prep.
        const v2f bb = *(const v2f*)&Wk[bvoff + g * 256 + kk * 64];
        c = __builtin_amdgcn_wmma_f32_16x16x4_f32(
                /*neg_a=*/false, a, /*neg_b=*/false, bb,
                /*c_mod=*/(short)0, c, /*reuse_a=*/false, /*reuse_b=*/false);
      }
      acc[g] = c;
    }

    __syncthreads();  // every wave finished reading this pass's inputs

    // ---- write phase: D rows v (+8 for hi lanes), col = 16*(wave*16+g)+l16.
#pragma unroll
    for (int g = 0; g < 16; ++g) {
#pragma unroll
      for (int v = 0; v < 8; ++v) {
        lds[dbase[v] + g * 16] = acc[g][v];
      }
    }
    __syncthreads();
  }

  // ---- store 16 rows back ---------------------------------------------------
  float4* dst = (float4*)(out + rowBase * NCOL);
  for (int i = tid; i < ROWS_PER_WG * NCOL / 4; i += THREADS) {
    const int r  = i >> 10;
    const int c4 = i & 1023;
    dst[r * 1024 + c4] = *(const float4*)&lds[r * LDS_STRIDE + c4 * 4];
  }
}

// ---------------------------------------------------------------------------
extern "C" void kernel_launch(void* const* d_in, const int* in_sizes, int n_in,
                              void* d_out, int out_size, void* d_ws, size_t ws_size,
                              hipStream_t stream) {
  const float* X      = (const float*)d_in[0];   // (8192, 4096) f32
  const float* params = (const float*)d_in[1];   // (2048, 24) f32
  float* out = (float*)d_out;                    // (8192, 4096) f32
  float* W   = (float*)d_ws;                     // 6*65536 floats = 1.5 MB scratch

  const size_t ldsBytes = (size_t)ROWS_PER_WG * LDS_STRIDE * sizeof(float);  // 262400 B

  butterfly_prep<<<(NMACRO * NGROUP * 16) / 256, 256, 0, stream>>>(params, W);
  butterfly_main<<<BATCH / ROWS_PER_WG, THREADS, ldsBytes, stream>>>(X, W, out);
}